// ParabolicIntegrate_2d_77592879169926
// MI455X (gfx1250) — compile-verified
//
#include <hip/hip_runtime.h>

// ---------------------------------------------------------------------------
// Parabolic SPDE integration pipeline for MI455X (gfx1250, wave32).
//
// laplace(u) = C*(P*u + u*P) with corner zeroing, P = 64x64 periodic
// tridiagonal second-difference matrix; both matmuls on
// V_WMMA_F32_16X16X4_F32 (12 WMMA per 16x16 tile using P's block-tridiagonal
// structure). All P fragments are built once per wave and pinned in VGPRs via
// empty inline-asm (blocks rematerialization); scan state lives in registers
// (C-layout owner lanes) with an LDS mirror for cross-tile fragment loads.
// Steady-state step per tile: 12 DS fragment loads + 12 WMMA + FMA update.
// ---------------------------------------------------------------------------

#define BB 32
#define TT 64
#define NX 64
#define NY 64
#define SP 68            // LDS row stride in floats
#define DT_F 0.2f
#define INV_DT 5.0f
#define CC 0.2f          // DT/DX^2
#define INV_DX 1.0f
#define TSLICE (NX * NY)        // 4096  floats per [X,Y] slice
#define OSLICE (NX * NY * 7)    // 28672 floats per output time slice

typedef float v2f __attribute__((ext_vector_type(2)));
typedef float v8f __attribute__((ext_vector_type(8)));

static __device__ __forceinline__ v8f wmma_f32_16x16x4(v2f a, v2f b, v8f c) {
#if defined(__HIP_DEVICE_COMPILE__)
  return __builtin_amdgcn_wmma_f32_16x16x4_f32(false, a, false, b, (short)0, c,
                                               false, false);
#else
  (void)a; (void)b;
  return c;
#endif
}

// Pin a value into a VGPR: unknown provenance => cannot be rematerialized.
static __device__ __forceinline__ float opq(float x) {
#if defined(__HIP_DEVICE_COMPILE__)
  asm volatile("" : "+v"(x));
#endif
  return x;
}

static __device__ __forceinline__ float pval(int a, int b) {
  int d = a - b;
  d = d < 0 ? -d : d;
  return d == 0 ? -2.0f : ((d == 1 || d == 63) ? 1.0f : 0.0f);
}

// Fragment layouts (ISA 7.12.2): A 16x4: lane m=lane&15, K = vgpr + 2*(lane>>4).
// B 4x16 (mirror assumption): lane n=lane&15, K row = vgpr + 2*(lane>>4).
// C/D: vgpr r -> row r (lanes 0-15) / r+8 (lanes 16-31), col = lane&15.

// x-part P A-fragments: depend only on row-block i (shared by a wave's tiles).
struct PaCtx { v2f pa[6]; };

static __device__ __forceinline__ PaCtx make_pa(int i, int m, int h2) {
  PaCtx p;
  const int ib = i * 16;
#pragma unroll
  for (int e = 0; e < 4; ++e) {                    // diag K-tile kt == i
    const int kb = ib + e * 4 + h2;
    p.pa[e].x = opq(pval(ib + m, kb));
    p.pa[e].y = opq(pval(ib + m, kb + 1));
  }
  {                                                // kt=(i+1)&3, ks=0
    const int kb = ((i + 1) & 3) * 16 + h2;
    p.pa[4].x = opq(pval(ib + m, kb));
    p.pa[4].y = opq(pval(ib + m, kb + 1));
  }
  {                                                // kt=(i+3)&3, ks=3
    const int kb = ((i + 3) & 3) * 16 + 12 + h2;
    p.pa[5].x = opq(pval(ib + m, kb));
    p.pa[5].y = opq(pval(ib + m, kb + 1));
  }
  return p;
}

// Per-tile context: y-part P B-fragments, LDS offsets, per-point constants.
struct TileCtx {
  v2f pb[6];
  int bx, xo4, xo5;   // x-part u(B) bases (pair at +SP)
  int by, yo4, yo5;   // y-part u(A) bases (pair at +1, 8B aligned)
  float ccm[8];       // CC * corner-mask
  int lidx[8];        // LDS store index of this lane's C-layout points
  int rel[8];         // row*64+col of this lane's C-layout points
};

static __device__ __forceinline__ TileCtx make_tile(int i, int j, int m,
                                                    int h2, int h8) {
  TileCtx c;
  const int ib = i * 16, jb = j * 16;
#pragma unroll
  for (int e = 0; e < 4; ++e) {                    // diag K-tile kt == j
    const int kb = jb + e * 4 + h2;
    c.pb[e].x = opq(pval(kb, jb + m));
    c.pb[e].y = opq(pval(kb + 1, jb + m));
  }
  {                                                // kt=(j+1)&3, ks=0
    const int kb = ((j + 1) & 3) * 16 + h2;
    c.pb[4].x = opq(pval(kb, jb + m));
    c.pb[4].y = opq(pval(kb + 1, jb + m));
    c.yo4 = (ib + m) * SP + kb;
  }
  {                                                // kt=(j+3)&3, ks=3
    const int kb = ((j + 3) & 3) * 16 + 12 + h2;
    c.pb[5].x = opq(pval(kb, jb + m));
    c.pb[5].y = opq(pval(kb + 1, jb + m));
    c.yo5 = (ib + m) * SP + kb;
  }
  c.bx = (ib + h2) * SP + jb + m;
  c.xo4 = (((i + 1) & 3) * 16 + h2) * SP + jb + m;
  c.xo5 = (((i + 3) & 3) * 16 + 12 + h2) * SP + jb + m;
  c.by = (ib + m) * SP + jb + h2;
#pragma unroll
  for (int r = 0; r < 8; ++r) {
    const int row = ib + r + h8;
    const int col = jb + m;
    const bool corner =
        (row == 0 || row == NX - 1) && (col == 0 || col == NY - 1);
    c.ccm[r] = opq(corner ? 0.f : CC);
    c.lidx[r] = row * SP + col;
    c.rel[r] = row * NY + col;
  }
  return c;
}

static __device__ __forceinline__ v8f lap(const float* __restrict__ su,
                                          const PaCtx& p, const TileCtx& c) {
  v8f acc = {0.f, 0.f, 0.f, 0.f, 0.f, 0.f, 0.f, 0.f};
  // x-part: P * u  (u as B; rows SP apart)
#pragma unroll
  for (int e = 0; e < 4; ++e) {
    v2f b;
    b.x = su[c.bx + e * 4 * SP];
    b.y = su[c.bx + e * 4 * SP + SP];
    acc = wmma_f32_16x16x4(p.pa[e], b, acc);
  }
  {
    v2f b; b.x = su[c.xo4]; b.y = su[c.xo4 + SP];
    acc = wmma_f32_16x16x4(p.pa[4], b, acc);
  }
  {
    v2f b; b.x = su[c.xo5]; b.y = su[c.xo5 + SP];
    acc = wmma_f32_16x16x4(p.pa[5], b, acc);
  }
  // y-part: u * P  (u as A; consecutive pair, 8B aligned)
#pragma unroll
  for (int e = 0; e < 4; ++e) {
    v2f a = *reinterpret_cast<const v2f*>(su + c.by + e * 4);
    acc = wmma_f32_16x16x4(a, c.pb[e], acc);
  }
  {
    v2f a = *reinterpret_cast<const v2f*>(su + c.yo4);
    acc = wmma_f32_16x16x4(a, c.pb[4], acc);
  }
  {
    v2f a = *reinterpret_cast<const v2f*>(su + c.yo5);
    acc = wmma_f32_16x16x4(a, c.pb[5], acc);
  }
  return acc;
}

// ---------------- Kernel 1: dW (ch0) and I_xi = integrate(dW) (ch1) ---------
__global__ __launch_bounds__(256) void k1_pass1(const float* __restrict__ W,
                                                float* __restrict__ out) {
  __shared__ float su[NX * SP];
  const int b = blockIdx.x;
  const int tid = (int)threadIdx.x;
  const int lane = tid & 31;
  const int wave = tid >> 5;
  const int m = lane & 15;
  const int h2 = (lane >> 4) << 1;
  const int h8 = (lane >> 4) << 3;

  for (int idx = tid; idx < NX * SP; idx += 256) su[idx] = 0.f;

  const int i = wave >> 1;
  const int j0 = (wave & 1) * 2, j1 = j0 + 1;
  const PaCtx pa = make_pa(i, m, h2);
  const TileCtx c0 = make_tile(i, j0, m, h2, h8);
  const TileCtx c1 = make_tile(i, j1, m, h2, h8);

  const float* __restrict__ wb = W + (long)b * TT * TSLICE;
  float* __restrict__ ob = out + (long)b * TT * OSLICE;

  float ureg[2][8];               // scan state: this lane's C-layout points
  float wm1[2][8], wm2[2][8];     // W[t-1], W[t-2]
#pragma unroll
  for (int q = 0; q < 2; ++q) {
    const TileCtx& c = q ? c1 : c0;
#pragma unroll
    for (int r = 0; r < 8; ++r) {
      ureg[q][r] = 0.f;
      wm1[q][r] = wb[c.rel[r]];        // W at t=0
      wm2[q][r] = 0.f;
      ob[c.rel[r] * 7 + 0] = 0.f;      // t=0 slices
      ob[c.rel[r] * 7 + 1] = 0.f;
    }
  }
  __syncthreads();

  for (int t = 1; t < TT; ++t) {
    v8f acc0 = lap(su, pa, c0);
    v8f acc1 = lap(su, pa, c1);
    __syncthreads();   // all WMMA reads of u done before anyone updates u
    const int wt = t * TSLICE;
    const int ot = t * OSLICE;
#pragma unroll
    for (int q = 0; q < 2; ++q) {
      const TileCtx& c = q ? c1 : c0;
      const v8f acc = q ? acc1 : acc0;
#pragma unroll
      for (int r = 0; r < 8; ++r) {
        const float w_t = wb[wt + c.rel[r]];
        const float dw_t = (w_t - wm1[q][r]) * INV_DT;       // dW[t]
        float force = 0.f;                                   // dW[t-1]*DT
        if (t >= 2) force = ((wm1[q][r] - wm2[q][r]) * INV_DT) * DT_F;
        wm2[q][r] = wm1[q][r];
        wm1[q][r] = w_t;
        const float un = ureg[q][r] + c.ccm[r] * acc[r] + force;
        ureg[q][r] = un;
        su[c.lidx[r]] = un;
        ob[ot + c.rel[r] * 7 + 0] = dw_t;
        ob[ot + c.rel[r] * 7 + 1] = un;
        if (r == 0 && t < TT - 1)
          __builtin_prefetch(&wb[wt + TSLICE + c.rel[0]], 0, 3);
      }
    }
    __syncthreads();
  }
}

// -------- Kernel 2: f4/f5/f6 scans, one (batch, field) per block ------------
template <int FIELD>
static __device__ __forceinline__ void k2_body(float* __restrict__ out) {
  __shared__ float su[NX * SP];
  const int b = blockIdx.x;
  const int tid = (int)threadIdx.x;
  const int lane = tid & 31;
  const int wave = tid >> 5;
  const int m = lane & 15;
  const int h2 = (lane >> 4) << 1;
  const int h8 = (lane >> 4) << 3;

  for (int idx = tid; idx < NX * SP; idx += 256) su[idx] = 0.f;

  const int i = wave >> 1;
  const int j0 = (wave & 1) * 2, j1 = j0 + 1;
  const PaCtx pa = make_pa(i, m, h2);
  const TileCtx c0 = make_tile(i, j0, m, h2, h8);
  const TileCtx c1 = make_tile(i, j1, m, h2, h8);

  float* __restrict__ ob = out + (long)b * TT * OSLICE;

  float ureg[2][8];
  int rsh7[2][8];    // *7 offsets of x-shifted points (f6's I2 factor)
#pragma unroll
  for (int q = 0; q < 2; ++q) {
    const TileCtx& c = q ? c1 : c0;
#pragma unroll
    for (int r = 0; r < 8; ++r) {
      ureg[q][r] = 0.f;
      const int row = c.rel[r] >> 6;
      const int col = c.rel[r] & 63;
      const int xs = (row >= 1 && row <= NX - 2) ? (row - 1) : (NX - 2);
      rsh7[q][r] = (xs * NY + col) * 7;
      ob[c.rel[r] * 7 + 4 + FIELD] = 0.f;          // t = 0 slice
    }
  }
  __syncthreads();

  for (int t = 1; t < TT; ++t) {
    v8f acc0 = lap(su, pa, c0);
    v8f acc1 = lap(su, pa, c1);
    __syncthreads();
    const int ot = t * OSLICE;
    const int om1 = (t - 1) * OSLICE;
    const int ota = ((t >= 2) ? (t - 2) : (TT - 2)) * OSLICE;
#pragma unroll
    for (int q = 0; q < 2; ++q) {
      const TileCtx& c = q ? c1 : c0;
      const v8f acc = q ? acc1 : acc0;
#pragma unroll
      for (int r = 0; r < 8; ++r) {
        const int r7 = c.rel[r] * 7;
        float force;
        if (FIELD == 0) {            // f4: dW[t-1] * I_xi[t-1]
          force = DT_F * ob[om1 + r7 + 0] * ob[om1 + r7 + 1];
        } else if (FIELD == 1) {     // f5: I_xi[t-1]^2
          const float Ii = ob[om1 + r7 + 1];
          force = DT_F * Ii * Ii;
        } else {                     // f6: I1[t-1] * I2[t-1]
          const float i1v = -ob[ota + r7 + 1] * INV_DX;
          const float i2v = -ob[om1 + rsh7[q][r] + 1] * INV_DX;
          force = DT_F * i1v * i2v;
        }
        const float un = ureg[q][r] + c.ccm[r] * acc[r] + force;
        ureg[q][r] = un;
        su[c.lidx[r]] = un;
        ob[ot + r7 + 4 + FIELD] = un;
      }
    }
    __syncthreads();
  }
}

__global__ __launch_bounds__(256) void k2_pass2(float* __restrict__ out) {
  switch (blockIdx.y) {
    case 0: k2_body<0>(out); break;
    case 1: k2_body<1>(out); break;
    default: k2_body<2>(out); break;
  }
}

// -------- Kernel 3: pointwise derivatives I1 (ch2), I2 (ch3) of I_xi --------
__global__ __launch_bounds__(256) void k3_deriv(float* __restrict__ out) {
  const long gid = (long)blockIdx.x * 256 + threadIdx.x;  // over B*T*X*Y
  const int y = (int)(gid & 63);
  const int x = (int)((gid >> 6) & 63);
  const int t = (int)((gid >> 12) & 63);
  const int b = (int)(gid >> 18);
  const int s  = (t == 0 || t == TT - 1) ? (TT - 2) : (t - 1);
  const int xs = (x == 0 || x == NX - 1) ? (NX - 2) : (x - 1);
  const long base = (long)b * TT * OSLICE;
  const float i1v = -out[base + (long)s * OSLICE + (x * NY + y) * 7 + 1] * INV_DX;
  const float i2v = -out[base + (long)t * OSLICE + (xs * NY + y) * 7 + 1] * INV_DX;
  const long o = gid * 7;
  out[o + 2] = i1v;
  out[o + 3] = i2v;
}

extern "C" void kernel_launch(void* const* d_in, const int* in_sizes, int n_in,
                              void* d_out, int out_size, void* d_ws,
                              size_t ws_size, hipStream_t stream) {
  (void)in_sizes; (void)n_in; (void)out_size; (void)d_ws; (void)ws_size;
  const float* W = (const float*)d_in[0];
  float* out = (float*)d_out;

  k1_pass1<<<dim3(BB), dim3(256), 0, stream>>>(W, out);
  k2_pass2<<<dim3(BB, 3), dim3(256), 0, stream>>>(out);
  k3_deriv<<<dim3((BB * TT * NX * NY) / 256), dim3(256), 0, stream>>>(out);
}